// Set2SetLoss_54760833024347
// MI455X (gfx1250) — compile-verified
//
#include <hip/hip_runtime.h>
#include <hip/hip_bf16.h>
#include <math.h>

typedef __attribute__((ext_vector_type(2))) float v2f;
typedef __attribute__((ext_vector_type(8))) float v8f;

#define NPT 128
#define DDIM 5
#define FLT_BIG 3.402823466e38f
static __device__ __constant__ float PI_F = 3.14159265358979323846f;

// One wave32 per block; one block per batch.
__global__ __launch_bounds__(32) void s2s_batch_kernel(const float* __restrict__ input,
                                                       const float* __restrict__ target,
                                                       float* __restrict__ partial) {
    __shared__ float Clds[NPT * NPT];          // 64 KB cost matrix
    __shared__ float nx_s[NPT], ny_s[NPT];     // squared norms
    __shared__ float u_sm[NPT + 1], v_sm[NPT + 1];
    __shared__ int   p_sm[NPT + 1];

    const int b    = blockIdx.x;
    const int lane = threadIdx.x;              // 0..31 (wave32)
    const float* X = input  + (size_t)b * NPT * DDIM;
    const float* Y = target + (size_t)b * NPT * DDIM;

    // ---------- squared norms ----------
    #pragma unroll
    for (int k = 0; k < 4; ++k) {
        int i = lane + 32 * k;
        float sx = 0.f, sy = 0.f;
        #pragma unroll
        for (int d = 0; d < DDIM; ++d) {
            float a = X[i * DDIM + d]; sx += a * a;
            float c = Y[i * DDIM + d]; sy += c * c;
        }
        nx_s[i] = sx; ny_s[i] = sy;
    }
    asm volatile("s_wait_dscnt 0x0" ::: "memory");

    // ---------- cost matrix via V_WMMA_F32_16X16X4_F32 ----------
    // G = X * Y^T (K = 5 padded to 8 -> two chained K=4 WMMAs per 16x16 tile)
    // A 16x4 f32 layout: lanes 0-15 hold M=0..15 with VGPR0=K0,VGPR1=K1;
    //                    lanes 16-31 hold M=0..15 with VGPR0=K2,VGPR1=K3.
    const int m     = lane & 15;
    const int khalf = lane >> 4;
    for (int ti = 0; ti < 8; ++ti) {
        const int arow = ti * 16 + m;
        v2f a1, a2;
        a1.x = X[arow * DDIM + 2 * khalf + 0];
        a1.y = X[arow * DDIM + 2 * khalf + 1];
        a2.x = (khalf == 0) ? X[arow * DDIM + 4] : 0.0f;   // d=4 (d=5..7 are zero pad)
        a2.y = 0.0f;
        for (int tj = 0; tj < 8; ++tj) {
            const int bcol = tj * 16 + m;
            v2f b1, b2;
            b1.x = Y[bcol * DDIM + 2 * khalf + 0];
            b1.y = Y[bcol * DDIM + 2 * khalf + 1];
            b2.x = (khalf == 0) ? Y[bcol * DDIM + 4] : 0.0f;
            b2.y = 0.0f;
            v8f c = {};
            c = __builtin_amdgcn_wmma_f32_16x16x4_f32(false, a1, false, b1,
                                                      (short)0, c, false, false);
            c = __builtin_amdgcn_wmma_f32_16x16x4_f32(false, a2, false, b2,
                                                      (short)0, c, false, false);
            // C/D 16x16 f32: VGPR r -> row (r + 8*khalf), col = lane&15
            #pragma unroll
            for (int r = 0; r < 8; ++r) {
                int orow = ti * 16 + khalf * 8 + r;
                int ocol = tj * 16 + m;
                float g  = c[r];
                float d2 = nx_s[orow] + ny_s[ocol] - 2.0f * g;
                Clds[orow * NPT + ocol] = sqrtf(fmaxf(d2, 0.0f));
            }
        }
    }

    // ---------- init Hungarian state ----------
    #pragma unroll
    for (int k = 0; k < 5; ++k) {
        int j = lane + 32 * k;
        if (j <= NPT) { u_sm[j] = 0.f; v_sm[j] = 0.f; p_sm[j] = 0; }
    }
    asm volatile("s_wait_dscnt 0x0" ::: "memory");

    volatile float* u = u_sm;
    volatile float* v = v_sm;
    volatile int*   p = p_sm;

    // Each lane owns columns j = 1 + lane + 32*k, k = 0..3 (1-based columns).
    for (int i = 1; i <= NPT; ++i) {
        if (lane == 0) p[0] = i;
        float minvk[4]; int wayk[4]; int usedk[4];
        #pragma unroll
        for (int k = 0; k < 4; ++k) { minvk[k] = FLT_BIG; wayk[k] = 0; usedk[k] = 0; }
        int j0 = 0;

        while (true) {
            // mark column j0 used (col 0 handled implicitly by lane 0 below)
            #pragma unroll
            for (int k = 0; k < 4; ++k)
                if (lane + 32 * k + 1 == j0) usedk[k] = 1;

            const int   i0  = p[j0];
            const float ui0 = u[i0];

            #pragma unroll
            for (int k = 0; k < 4; ++k) {
                if (!usedk[k]) {
                    int j = lane + 32 * k + 1;
                    float cur = Clds[i0 * NPT + (j - 1)] - ui0 - v[j];
                    if (cur < minvk[k]) { minvk[k] = cur; wayk[k] = j0; }
                }
            }

            // argmin over unused columns (lowest index on ties, matches np.argmin)
            float bv = FLT_BIG; int bj = NPT + 1;
            #pragma unroll
            for (int k = 0; k < 4; ++k) {
                int j = lane + 32 * k + 1;
                if (!usedk[k] && (minvk[k] < bv || (minvk[k] == bv && j < bj))) {
                    bv = minvk[k]; bj = j;
                }
            }
            #pragma unroll
            for (int off = 16; off > 0; off >>= 1) {
                float ov = __shfl_xor(bv, off, 32);
                int   oj = __shfl_xor(bj, off, 32);
                if (ov < bv || (ov == bv && oj < bj)) { bv = ov; bj = oj; }
            }
            const float delta = bv;
            const int   j1    = bj;

            // u[p[used]] += delta; v[used] -= delta; minv[~used] -= delta
            #pragma unroll
            for (int k = 0; k < 4; ++k) {
                int j = lane + 32 * k + 1;
                if (usedk[k]) {
                    v[j] = v[j] - delta;
                    int pj = p[j];               // distinct rows across used cols
                    u[pj] = u[pj] + delta;
                } else {
                    minvk[k] -= delta;
                }
            }
            if (lane == 0) { int p0 = p[0]; u[p0] = u[p0] + delta; }  // column 0

            j0 = j1;
            if (p[j0] == 0) break;
        }

        // augmenting-path update: while j0: j1 = way[j0]; p[j0] = p[j1]; j0 = j1
        while (j0 != 0) {
            int wv = 0;
            #pragma unroll
            for (int k = 0; k < 4; ++k)
                if (lane + 32 * k + 1 == j0) wv = wayk[k];
            int w  = __shfl(wv, (j0 - 1) & 31, 32);
            int pw = p[w];
            if (lane == 0) p[j0] = pw;
            asm volatile("s_wait_dscnt 0x0" ::: "memory");
            j0 = w;
        }
    }

    asm volatile("s_wait_dscnt 0x0" ::: "memory");

    // ---------- per-batch loss partials over matched pairs (row p[j]-1, col j-1) ----------
    float s[5] = {0.f, 0.f, 0.f, 0.f, 0.f};
    #pragma unroll
    for (int k = 0; k < 4; ++k) {
        int jj = lane + 32 * k;          // 0-based column
        int r  = p[jj + 1] - 1;          // matched row
        #pragma unroll
        for (int d = 0; d < DDIM; ++d) {
            float dd = X[r * DDIM + d] - Y[jj * DDIM + d];
            if (d == 4) {                // wrap to [-pi, pi), jnp.remainder semantics
                float t = fmodf(dd + PI_F, 2.0f * PI_F);
                if (t < 0.0f) t += 2.0f * PI_F;
                dd = t - PI_F;
            }
            s[d] += dd * dd;
        }
    }
    #pragma unroll
    for (int d = 0; d < DDIM; ++d)
        #pragma unroll
        for (int off = 16; off > 0; off >>= 1)
            s[d] += __shfl_xor(s[d], off, 32);

    if (lane == 0) {
        #pragma unroll
        for (int d = 0; d < DDIM; ++d) partial[b * DDIM + d] = s[d];
    }
}

// Single wave32: reduce 128 batches x 5 components -> [total, x, y, E, theta, phi]
__global__ __launch_bounds__(32) void s2s_reduce_kernel(const float* __restrict__ partial,
                                                        float* __restrict__ out) {
    const int lane = threadIdx.x;
    float comp[5];
    float total = 0.f;
    #pragma unroll
    for (int d = 0; d < DDIM; ++d) {
        float s = 0.f;
        #pragma unroll
        for (int k = 0; k < 4; ++k) s += partial[(lane + 32 * k) * DDIM + d];
        #pragma unroll
        for (int off = 16; off > 0; off >>= 1) s += __shfl_xor(s, off, 32);
        s *= (1.0f / (float)(NPT * NPT));   // mean over B*N
        comp[d] = s;
        total  += s;
    }
    if (lane == 0) {
        out[0] = total;
        #pragma unroll
        for (int d = 0; d < DDIM; ++d) out[1 + d] = comp[d];
    }
}

extern "C" void kernel_launch(void* const* d_in, const int* in_sizes, int n_in,
                              void* d_out, int out_size, void* d_ws, size_t ws_size,
                              hipStream_t stream) {
    const float* input  = (const float*)d_in[0];   // (128,128,5) f32
    const float* target = (const float*)d_in[1];   // (128,128,5) f32
    float* out     = (float*)d_out;                // 6 f32
    float* partial = (float*)d_ws;                 // 128*5 f32 partials

    s2s_batch_kernel<<<dim3(128), dim3(32), 0, stream>>>(input, target, partial);
    s2s_reduce_kernel<<<dim3(1), dim3(32), 0, stream>>>(partial, out);
}